// SampleLayer_11759620456883
// MI455X (gfx1250) — compile-verified
//
#include <hip/hip_runtime.h>
#include <hip/hip_bf16.h>
#include <cstdint>

#define VOCAB     128000
#define HIST_LEN  2048
#define TOPK      15u
#define NTHREADS  1024
#define NTILES    (VOCAB / NTHREADS)   // 125 exact

typedef unsigned int u32x4 __attribute__((ext_vector_type(4)));
typedef int          i32x8 __attribute__((ext_vector_type(8)));
typedef int          i32x4 __attribute__((ext_vector_type(4)));

// ---- monotonic float<->uint mapping (unsigned order == float order) ----
__device__ __forceinline__ unsigned mapf(float x) {
    unsigned b = __float_as_uint(x);
    return b ^ ((b & 0x80000000u) ? 0xFFFFFFFFu : 0x80000000u);
}
__device__ __forceinline__ float unmapf(unsigned u) {
    unsigned b = (u & 0x80000000u) ? (u ^ 0x80000000u) : ~u;
    return __uint_as_float(b);
}
__device__ __forceinline__ float penalize(float s) {
    return (s < 0.0f) ? s * 1.35f : s / 1.35f;
}

// ---- CDNA5 async global->LDS copy (ISA 15.18.3 op 96) + split wait counter ----
__device__ __forceinline__ void async_b32(unsigned lds_byte_addr, const void* gaddr) {
    asm volatile("global_load_async_to_lds_b32 %0, %1, off"
                 :: "v"(lds_byte_addr), "v"(gaddr) : "memory");
}
template <int N>
__device__ __forceinline__ void wait_async() {
    asm volatile("s_wait_asynccnt %0" :: "n"(N) : "memory");
}
__device__ __forceinline__ unsigned lds_off(const void* p) {
    // generic (flat) pointer to LDS: low 32 bits are the LDS byte offset
    return (unsigned)(uint64_t)(uintptr_t)p;
}

// ---- CDNA5 Tensor Data Mover: 1-D DMA of nelem 4-byte words into LDS ----
// D# per CDNA5 ISA: group0 = {count|flags, lds_addr, global_addr, type=2},
// group1 = {data_size=4B, tensor_dim0=nelem, tile_dim0=nelem, stride=nelem}.
// This toolchain exposes the 6-arg builtin: (v4u g0, v8i g1, v4i, v4i, v8i, i32 cpol).
__device__ __forceinline__ void tdm_load_1d_b32(unsigned lds_addr, const void* gaddr,
                                                unsigned nelem /* <= 65535 */) {
    unsigned long long ga = (unsigned long long)(uintptr_t)gaddr;
    u32x4 g0;
    g0.x = 1u;                                                   // count=1, user mode
    g0.y = lds_addr;                                             // bits[63:32]
    g0.z = (unsigned)(ga & 0xFFFFFFFFu);                         // global_addr[31:0]
    g0.w = (unsigned)((ga >> 32) & 0x01FFFFFFu) | (2u << 30);    // addr[56:32] | type=2
    i32x8 g1;
    g1[0] = (int)(2u << 16);                    // wg_mask=0, data_size=2 (4 bytes)
    g1[1] = (int)((nelem & 0xFFFFu) << 16);     // tensor_dim0[15:0] in bits[63:48]
    g1[2] = (int)((nelem >> 16) & 0xFFFFu);     // tensor_dim0[31:16]; tensor_dim1=0
    g1[3] = (int)((nelem & 0xFFFFu) << 16);     // tile_dim0 = nelem (bits[127:112])
    g1[4] = 0;                                  // tile_dim1=0, tile_dim2=0 (1-D tile)
    g1[5] = (int)nelem;                         // tensor_dim0_stride[31:0]
    g1[6] = 0;
    g1[7] = 0;
    i32x4 z4 = {0, 0, 0, 0};
    i32x8 z8 = {0, 0, 0, 0, 0, 0, 0, 0};
#if __has_builtin(__builtin_amdgcn_tensor_load_to_lds)
    __builtin_amdgcn_tensor_load_to_lds(g0, g1, z4, z4, z8, 0);
#else
    asm volatile("tensor_load_to_lds %0, %1" :: "s"(g0), "s"(g1) : "memory");
#endif
}
__device__ __forceinline__ void wait_tensor0() {
#if __has_builtin(__builtin_amdgcn_s_wait_tensorcnt)
    __builtin_amdgcn_s_wait_tensorcnt(0);
#else
    asm volatile("s_wait_tensorcnt 0" ::: "memory");
#endif
}

// ---- find bin where cumulative-from-top count first reaches K (64K bins in LDS) ----
// writes: scal[0] = bin index, scal[1] = count strictly above that bin
__device__ void select_bin(unsigned* hist, unsigned* scan, unsigned* scal,
                           unsigned K, int tid) {
    const int hi = 65535 - (tid << 6);       // descending 64-bin chunk per thread
    unsigned s = 0;
#pragma unroll 8
    for (int j = 0; j < 64; ++j) s += hist[hi - j];
    scan[tid] = s;
    __syncthreads();
    // Hillis-Steele inclusive scan over chunk sums (top-down order)
    for (int off = 1; off < NTHREADS; off <<= 1) {
        unsigned add = (tid >= off) ? scan[tid - off] : 0u;
        __syncthreads();
        scan[tid] += add;
        __syncthreads();
    }
    const unsigned incl = scan[tid];
    const unsigned excl = incl - s;
    if (excl < K && K <= incl) {             // exactly one thread
        unsigned cum = excl;
        for (int j = 0; j < 64; ++j) {
            unsigned c = hist[hi - j];
            cum += c;
            if (cum >= K) { scal[0] = (unsigned)(hi - j); scal[1] = cum - c; break; }
        }
    }
    __syncthreads();
}

__global__ void __launch_bounds__(NTHREADS)
sample_fused_kernel(const float* __restrict__ logits, const int* __restrict__ y,
                    const float* __restrict__ noise, float* __restrict__ out) {
    extern __shared__ unsigned smem[];
    unsigned* hist   = smem;                       // 65536 u32 (256 KB) 16-bit radix hist
    unsigned* bitmap = hist + 65536;               // 4000  u32 (128000-bit penalty set)
    unsigned* scan   = bitmap + 4000;              // 1024  u32
    float*    sVal   = (float*)(scan + 1024);      // 1024  f32
    int*      sIdx   = (int*)(sVal + 1024);        // 1024  i32
    unsigned* scal   = (unsigned*)(sIdx + 1024);   // 8     u32
    float*    stageL = (float*)(scal + 8);         // 2x1024 f32 double buffer (logits)
    float*    stageN = stageL + 2048;              // 2x1024 f32 double buffer (noise)

    const int tid = threadIdx.x;
    const unsigned offL = lds_off(stageL);
    const unsigned offN = lds_off(stageN);

    // ---- phase 0: TDM DMA of y[2048] into LDS (wave 0), overlapped with zeroing ----
    int* yLds = (int*)sVal;                        // 8 KB: sVal+sIdx region, free until phase 5
    if (tid < 32) tdm_load_1d_b32(lds_off(yLds), y, HIST_LEN);
    for (int j = tid; j < 65536; j += NTHREADS) hist[j] = 0u;
    for (int j = tid; j < 4000;  j += NTHREADS) bitmap[j] = 0u;
    if (tid < 32) wait_tensor0();
    __syncthreads();

    // ---- phase 1: build membership bitmap of history tokens (from LDS copy) ----
    for (int t = tid; t < HIST_LEN; t += NTHREADS) {
        int v = yLds[t];
        atomicOr(&bitmap[v >> 5], 1u << (v & 31));
    }
    __syncthreads();

    // ---- phase 2: stream logits (async->LDS pipeline), hi-16 histogram ----
    async_b32(offL + (unsigned)tid * 4u, logits + tid);
    for (int t = 0; t < NTILES; ++t) {
        const int buf = t & 1;
        const bool more = (t + 1 < NTILES);
        if (more)
            async_b32(offL + (unsigned)((buf ^ 1) * NTHREADS + tid) * 4u,
                      logits + (t + 1) * NTHREADS + tid);
        if (more) wait_async<1>(); else wait_async<0>();
        const int i = t * NTHREADS + tid;
        float x = stageL[buf * NTHREADS + tid];
        if ((bitmap[i >> 5] >> (i & 31)) & 1u) x = penalize(x);
        atomicAdd(&hist[mapf(x) >> 16], 1u);
    }
    __syncthreads();

    // ---- phase 3: select hi-16 bin of the 15th largest ----
    select_bin(hist, scan, scal, TOPK, tid);
    const unsigned hiBin = scal[0];
    const unsigned above = scal[1];
    __syncthreads();

    // ---- phase 4: re-zero hist; lo-16 histogram within hiBin ----
    for (int j = tid; j < 65536; j += NTHREADS) hist[j] = 0u;
    __syncthreads();
    async_b32(offL + (unsigned)tid * 4u, logits + tid);
    for (int t = 0; t < NTILES; ++t) {
        const int buf = t & 1;
        const bool more = (t + 1 < NTILES);
        if (more)
            async_b32(offL + (unsigned)((buf ^ 1) * NTHREADS + tid) * 4u,
                      logits + (t + 1) * NTHREADS + tid);
        if (more) wait_async<1>(); else wait_async<0>();
        const int i = t * NTHREADS + tid;
        float x = stageL[buf * NTHREADS + tid];
        if ((bitmap[i >> 5] >> (i & 31)) & 1u) x = penalize(x);
        const unsigned u = mapf(x);
        if ((u >> 16) == hiBin) atomicAdd(&hist[u & 0xFFFFu], 1u);
    }
    __syncthreads();
    select_bin(hist, scan, scal, TOPK - above, tid);
    const float pivot = unmapf((hiBin << 16) | scal[0]);   // exact 15th-largest value

    // ---- phase 5: stream logits+noise, masked gumbel-style argmax ----
    // argmax(softmax(l)/noise) == argmax((l>=pivot ? exp(l) : 0)/noise): the softmax
    // denominator and exp(-max) are uniform positive scales -> order-invariant.
    float bestV = -INFINITY;
    int   bestI = 0x7FFFFFFF;
    async_b32(offL + (unsigned)tid * 4u, logits + tid);
    async_b32(offN + (unsigned)tid * 4u, noise + tid);
    for (int t = 0; t < NTILES; ++t) {
        const int buf = t & 1;
        const bool more = (t + 1 < NTILES);
        if (more) {
            async_b32(offL + (unsigned)((buf ^ 1) * NTHREADS + tid) * 4u,
                      logits + (t + 1) * NTHREADS + tid);
            async_b32(offN + (unsigned)((buf ^ 1) * NTHREADS + tid) * 4u,
                      noise + (t + 1) * NTHREADS + tid);
        }
        if (more) wait_async<2>(); else wait_async<0>();
        const int i = t * NTHREADS + tid;
        float x = stageL[buf * NTHREADS + tid];
        if ((bitmap[i >> 5] >> (i & 31)) & 1u) x = penalize(x);
        const float nz = stageN[buf * NTHREADS + tid];
        float v = 0.0f;                              // masked prob is exactly 0
        if (x >= pivot) v = __expf(x) / nz;
        if (v > bestV || (v == bestV && i < bestI)) { bestV = v; bestI = i; }
    }
    // cross-thread argmax reduction, first-index tie-break
    sVal[tid] = bestV; sIdx[tid] = bestI;
    __syncthreads();
    for (int s = NTHREADS / 2; s > 0; s >>= 1) {
        if (tid < s) {
            float ov = sVal[tid + s]; int oi = sIdx[tid + s];
            if (ov > sVal[tid] || (ov == sVal[tid] && oi < sIdx[tid])) {
                sVal[tid] = ov; sIdx[tid] = oi;
            }
        }
        __syncthreads();
    }
    const int sample = sIdx[0];

    // ---- phase 6: outputs: [sample, y[0..2047], sample] as f32 ----
    if (tid == 0) { out[0] = (float)sample; out[1 + HIST_LEN] = (float)sample; }
    for (int t = tid; t < HIST_LEN; t += NTHREADS) out[1 + t] = (float)y[t];
}

extern "C" void kernel_launch(void* const* d_in, const int* in_sizes, int n_in,
                              void* d_out, int out_size, void* d_ws, size_t ws_size,
                              hipStream_t stream) {
    (void)in_sizes; (void)n_in; (void)out_size; (void)d_ws; (void)ws_size;
    const float* logits = (const float*)d_in[0];
    const int*   y      = (const int*)d_in[1];
    const float* noise  = (const float*)d_in[2];
    float*       out    = (float*)d_out;
    // 65536 + 4000 + 1024 + 1024 + 1024 + 8 + 4096 u32 words = 306848 B < 320 KB WGP LDS
    const size_t smem_bytes = (size_t)(65536 + 4000 + 1024 + 1024 + 1024 + 8 + 4096) * 4;
    sample_fused_kernel<<<1, NTHREADS, smem_bytes, stream>>>(logits, y, noise, out);
}